// GCNLayer_25374666784877
// MI455X (gfx1250) — compile-verified
//
#include <hip/hip_runtime.h>
#include <hip/hip_bf16.h>

typedef __attribute__((ext_vector_type(2))) float v2f;
typedef __attribute__((ext_vector_type(8))) float v8f;

#define HT_BITS   23
#define HT_SIZE   (1u << HT_BITS)           // 8M slots, 64 MB, ~0.4 load factor
#define HT_MASK   (HT_SIZE - 1u)
#define EMPTY_KEY 0xFFFFFFFFFFFFFFFFull
#define DIM       64

// ---------------------------------------------------------------------------
// Init: hash table to EMPTY, degrees to 0, edge counter to 0, output to 0.
// ---------------------------------------------------------------------------
__global__ void gcn_init_kernel(unsigned long long* __restrict__ ht, int htsize,
                                unsigned int* __restrict__ deg, int n,
                                unsigned int* __restrict__ count,
                                float* __restrict__ out, int outn) {
    int gid = blockIdx.x * blockDim.x + threadIdx.x;
    if (gid < htsize) ht[gid] = EMPTY_KEY;
    if (gid < n)      deg[gid] = 0u;
    if (gid == 0)     *count = 0u;
    if (gid < outn)   out[gid] = 0.0f;
}

// ---------------------------------------------------------------------------
// Build coalesced undirected edge set + self loops via 64-bit open addressing.
// Candidate t: [0,E) forward edge, [E,2E) reversed edge, [2E,2E+N) self loop.
// First insert of a key bumps deg[dst] and appends to the compacted edge list.
// ---------------------------------------------------------------------------
__global__ void gcn_build_kernel(const int* __restrict__ ei, int E, int N,
                                 unsigned long long* __restrict__ ht,
                                 unsigned int* __restrict__ deg,
                                 int* __restrict__ esrc, int* __restrict__ edst,
                                 unsigned int* __restrict__ count) {
    int t = blockIdx.x * blockDim.x + threadIdx.x;
    int M = 2 * E + N;
    if (t >= M) return;

    int s, d;
    if (t < E)          { s = ei[t];         d = ei[E + t];  }
    else if (t < 2 * E) { int u = t - E;  s = ei[E + u]; d = ei[u]; }
    else                { s = d = t - 2 * E; }

    unsigned long long key =
        (unsigned long long)s * (unsigned long long)N + (unsigned long long)d;

    // splitmix64 finalizer
    unsigned long long z = key + 0x9E3779B97F4A7C15ull;
    z = (z ^ (z >> 30)) * 0xBF58476D1CE4E5B9ull;
    z = (z ^ (z >> 27)) * 0x94D049BB133111EBull;
    z = z ^ (z >> 31);
    unsigned int slot = (unsigned int)z & HT_MASK;

    for (;;) {
        unsigned long long prev = atomicCAS(&ht[slot], EMPTY_KEY, key);
        if (prev == EMPTY_KEY) {                 // newly inserted (unique edge)
            atomicAdd(&deg[d], 1u);
            unsigned int pos = atomicAdd(count, 1u);
            esrc[pos] = s;
            edst[pos] = d;
            break;
        }
        if (prev == key) break;                  // duplicate — coalesced away
        slot = (slot + 1u) & HT_MASK;
    }
}

// ---------------------------------------------------------------------------
// inv_sqrt[i] = deg[i]^-0.5  (deg clamped to >= 1)
// ---------------------------------------------------------------------------
__global__ void gcn_invsqrt_kernel(const unsigned int* __restrict__ deg,
                                   float* __restrict__ inv, int n) {
    int i = blockIdx.x * blockDim.x + threadIdx.x;
    if (i >= n) return;
    unsigned int d = deg[i];
    if (d == 0u) d = 1u;
    inv[i] = rsqrtf((float)d);
}

// ---------------------------------------------------------------------------
// h = x @ W^T + b with V_WMMA_F32_16X16X4_F32 (fp32 matrix core).
// Block = 128 threads = 4 waves; block owns 16 rows, wave w owns cols 16w..16w+15.
// A (16x4 f32): lane&15 = row M, lane>=16 -> K offset +2, 2 floats per lane.
// B (4x16 f32): mirrored (lane&15 = col N).  C/D: 8 VGPRs, M = r + 8*(lane>=16).
// 16 WMMA issues per wave accumulate K = 64.
// ---------------------------------------------------------------------------
__global__ void gcn_gemm_wmma_kernel(const float* __restrict__ x,
                                     const float* __restrict__ W,
                                     const float* __restrict__ bias,
                                     float* __restrict__ h, int nrows) {
    const int lane    = threadIdx.x & 31;
    const int wave    = threadIdx.x >> 5;        // 0..3 -> column tile
    const int rowbase = blockIdx.x * 16;
    if (rowbase >= nrows) return;                // block-uniform, EXEC stays all-1s
    const int colbase = wave * 16;
    const int ln      = lane & 15;
    const int koff    = (lane >> 4) * 2;         // 0 or 2
    const int hi8     = (lane >> 4) * 8;

    const float* __restrict__ xrow = x + (size_t)(rowbase + ln) * DIM;
    const float* __restrict__ wrow = W + (size_t)(colbase + ln) * DIM;   // W is (out,in)

    v8f c = {};
#pragma unroll
    for (int k = 0; k < DIM; k += 4) {
        float2 af = *(const float2*)(xrow + k + koff);
        float2 bf = *(const float2*)(wrow + k + koff);
        v2f a; a.x = af.x; a.y = af.y;
        v2f b; b.x = bf.x; b.y = bf.y;
        // (neg_a, A, neg_b, B, c_mod, C, reuse_a, reuse_b)
        c = __builtin_amdgcn_wmma_f32_16x16x4_f32(false, a, false, b,
                                                  (short)0, c, false, false);
    }

    const float bv = bias[colbase + ln];
    float* __restrict__ outp = h + (size_t)(rowbase + hi8) * DIM + colbase + ln;
#pragma unroll
    for (int r = 0; r < 8; ++r) {
        outp[(size_t)r * DIM] = c[r] + bv;
    }
}

// ---------------------------------------------------------------------------
// Scatter: out[dst] += inv[src]*inv[dst] * h[src].  16 threads per edge,
// each handles 4 consecutive columns (float4 gather, 4 float atomics).
// All of h/out stay L2-resident (192 MB L2), so atomics are L2-local RMW.
// ---------------------------------------------------------------------------
__global__ void gcn_scatter_kernel(const int* __restrict__ esrc,
                                   const int* __restrict__ edst,
                                   const unsigned int* __restrict__ count,
                                   const float* __restrict__ inv,
                                   const float* __restrict__ h,
                                   float* __restrict__ out) {
    int gid = blockIdx.x * blockDim.x + threadIdx.x;
    int e   = gid >> 4;
    if (e >= (int)(*count)) return;
    int cg  = (gid & 15) * 4;

    int s = esrc[e];
    int d = edst[e];
    float w = inv[s] * inv[d];

    float4 hv = *(const float4*)(h + (size_t)s * DIM + cg);
    float* op = out + (size_t)d * DIM + cg;
    atomicAdd(op + 0, w * hv.x);
    atomicAdd(op + 1, w * hv.y);
    atomicAdd(op + 2, w * hv.z);
    atomicAdd(op + 3, w * hv.w);
}

// ---------------------------------------------------------------------------
extern "C" void kernel_launch(void* const* d_in, const int* in_sizes, int n_in,
                              void* d_out, int out_size, void* d_ws, size_t ws_size,
                              hipStream_t stream) {
    const float* x    = (const float*)d_in[0];
    const int*   ei   = (const int*)d_in[1];
    const float* W    = (const float*)d_in[2];
    const float* bias = (const float*)d_in[3];
    float*       out  = (float*)d_out;

    const int N = in_sizes[0] / DIM;   // 100000
    const int E = in_sizes[1] / 2;     // 1600000
    const int M = 2 * E + N;           // max candidate edges (3.3M)

    // --- workspace carve-up (256B aligned) ---
    char* ws = (char*)d_ws;
    size_t off = 0;
    auto carve = [&](size_t bytes) {
        char* p = ws + off;
        off += (bytes + 255) & ~(size_t)255;
        return p;
    };
    unsigned long long* ht   = (unsigned long long*)carve((size_t)HT_SIZE * 8);
    unsigned int*       deg  = (unsigned int*)carve((size_t)N * 4);
    float*              inv  = (float*)carve((size_t)N * 4);
    int*                esrc = (int*)carve((size_t)M * 4);
    int*                edst = (int*)carve((size_t)M * 4);
    unsigned int*       cnt  = (unsigned int*)carve(256);
    float*              h    = (float*)carve((size_t)N * DIM * 4);
    (void)ws_size;

    const int outn = N * DIM;

    // 1) init
    {
        int maxe = (int)HT_SIZE;            // covers outn (6.4M) and N too? no:
        if (outn > maxe) maxe = outn;       // be safe
        int blocks = (maxe + 255) / 256;
        gcn_init_kernel<<<blocks, 256, 0, stream>>>(ht, (int)HT_SIZE, deg, N, cnt,
                                                    out, outn);
    }
    // 2) build coalesced edge set
    {
        int blocks = (M + 255) / 256;
        gcn_build_kernel<<<blocks, 256, 0, stream>>>(ei, E, N, ht, deg, esrc, edst, cnt);
    }
    // 3) degree^-1/2
    {
        int blocks = (N + 255) / 256;
        gcn_invsqrt_kernel<<<blocks, 256, 0, stream>>>(deg, inv, N);
    }
    // 4) h = x @ W^T + b   (WMMA fp32)
    {
        int blocks = (N + 15) / 16;         // 6250, N divisible by 16
        gcn_gemm_wmma_kernel<<<blocks, 128, 0, stream>>>(x, W, bias, h, N);
    }
    // 5) scatter-add aggregation
    {
        long long threads = (long long)M * 16;
        int blocks = (int)((threads + 255) / 256);
        gcn_scatter_kernel<<<blocks, 256, 0, stream>>>(esrc, edst, cnt, inv, h, out);
    }
}